// GCN_69226282877030
// MI455X (gfx1250) — compile-verified
//
#include <hip/hip_runtime.h>
#include <hip/hip_bf16.h>

typedef float v2f __attribute__((ext_vector_type(2)));
typedef float v8f __attribute__((ext_vector_type(8)));

// ---------------- CSR build kernels ----------------

__global__ void gcn_zero_i(int* __restrict__ p, int n) {
  int i = blockIdx.x * blockDim.x + threadIdx.x;
  if (i < n) p[i] = 0;
}

__global__ void gcn_count(const int* __restrict__ dst, int* __restrict__ cnt, int e) {
  int i = blockIdx.x * blockDim.x + threadIdx.x;
  if (i < e) atomicAdd(&cnt[dst[i]], 1);
}

// exclusive prefix scan of cnt[0..n) -> row_start[0..n], plus cursor copy. One block, 1024 thr.
__global__ __launch_bounds__(1024) void gcn_scan(const int* __restrict__ cnt,
                                                 int* __restrict__ row_start,
                                                 int* __restrict__ cursor, int n) {
  __shared__ int part[1024];
  const int tid = threadIdx.x;
  const int chunk = (n + 1023) / 1024;
  const int begin = tid * chunk;
  const int end = (begin + chunk < n) ? begin + chunk : n;
  int s = 0;
  for (int i = begin; i < end; i++) s += cnt[i];
  part[tid] = s;
  __syncthreads();
  for (int off = 1; off < 1024; off <<= 1) {       // Hillis-Steele inclusive scan
    int v = part[tid];
    int add = (tid >= off) ? part[tid - off] : 0;
    __syncthreads();
    part[tid] = v + add;
    __syncthreads();
  }
  int run = (tid > 0) ? part[tid - 1] : 0;         // exclusive base for this chunk
  for (int i = begin; i < end; i++) {
    row_start[i] = run;
    cursor[i] = run;
    run += cnt[i];
  }
  if (tid == 0) row_start[n] = part[1023];
}

__global__ void gcn_fill(const int* __restrict__ src, const int* __restrict__ dst,
                         int* __restrict__ cursor, int* __restrict__ csr, int e) {
  int i = blockIdx.x * blockDim.x + threadIdx.x;
  if (i < e) {
    int pos = atomicAdd(&cursor[dst[i]], 1);
    csr[pos] = src[i];
  }
}

// sort each node's adjacency by src id -> bitwise-deterministic accumulation order
__global__ void gcn_sort(int* __restrict__ csr, const int* __restrict__ row_start, int n) {
  int i = blockIdx.x * blockDim.x + threadIdx.x;
  if (i >= n) return;
  int b = row_start[i], e = row_start[i + 1];
  for (int j = b + 1; j < e; j++) {
    int v = csr[j];
    int k = j - 1;
    while (k >= b && csr[k] > v) { csr[k + 1] = csr[k]; k--; }
    csr[k + 1] = v;
  }
}

__global__ void gcn_dinv(const int* __restrict__ cnt, float* __restrict__ dinv, int n) {
  int i = blockIdx.x * blockDim.x + threadIdx.x;
  if (i < n) dinv[i] = rsqrtf((float)cnt[i] + 1.0f);   // deg = in-degree + 1 (self loop)
}

// ---------------- fp32 WMMA GEMM: Out[N,FOUT] = act(A)[N,128] @ W[128,FOUT] ----------------
// FOUT/16 waves per block; B fragments hoisted into registers; grid-stride over row tiles.
// BN=true fuses the previous layer's BN scale/shift + ReLU into the LDS staging of A.
template<int FOUT, bool BN>
__global__ __launch_bounds__(FOUT * 2)
void gcn_gemm_wmma(const float* __restrict__ A, const float* __restrict__ W,
                   const float* __restrict__ coef, float* __restrict__ Out,
                   int nrows, int ntiles) {
  __shared__ float sA[16][132];                 // pad 128->132: conflict-free column reads
  const int tid  = threadIdx.x;
  const int nthr = FOUT * 2;
  const int wave = tid >> 5;
  const int lane = tid & 31;
  const int lrow = lane & 15;
  const int khi  = (lane >> 4) << 1;            // lanes 16-31 hold K+2,K+3
  const int col  = wave * 16 + lrow;

  v2f bf[32];                                   // all B fragments for K=128, loaded once
#pragma unroll
  for (int kk = 0; kk < 32; kk++) {
    bf[kk].x = W[(size_t)(4 * kk + khi)     * FOUT + col];
    bf[kk].y = W[(size_t)(4 * kk + khi + 1) * FOUT + col];
  }

  for (int tile = blockIdx.x; tile < ntiles; tile += gridDim.x) {
    const int row0 = tile * 16;
    __syncthreads();                            // previous tile's readers done
    for (int i = tid; i < 16 * 128; i += nthr) {
      int r = i >> 7, cc = i & 127;
      int gr = row0 + r;
      float v = 0.0f;
      if (gr < nrows) {
        v = A[(size_t)gr * 128 + cc];
        if (BN) {                               // fused BN scale/shift + ReLU
          v = v * coef[cc] + coef[128 + cc];
          v = v > 0.0f ? v : 0.0f;
        }
      }
      sA[r][cc] = v;
    }
    __syncthreads();

    v8f acc = {};
#pragma unroll
    for (int kk = 0; kk < 32; kk++) {
      v2f a;
      a.x = sA[lrow][4 * kk + khi];
      a.y = sA[lrow][4 * kk + khi + 1];
      acc = __builtin_amdgcn_wmma_f32_16x16x4_f32(
          false, a, false, bf[kk], (short)0, acc, false, false);
    }

    const int rbase = row0 + ((lane >> 4) << 3);
#pragma unroll
    for (int r = 0; r < 8; r++) {
      int gr = rbase + r;
      if (gr < nrows) Out[(size_t)gr * FOUT + col] = acc[r];
    }
  }
}

// ---------------- CSR gather conv: out[d] = dinv[d]*sum_s t[s]*dinv[s] + t[d]*dinv[d]^2 + b ----
// One wave32 per node; each node row is a contiguous 512B/256B burst -> zero atomics.
template<int F>
__global__ void gcn_gather(const float* __restrict__ t, const float* __restrict__ dinv,
                           const int* __restrict__ csr, const int* __restrict__ row_start,
                           const float* __restrict__ bias, float* __restrict__ out, int n) {
  int wid  = (blockIdx.x * blockDim.x + threadIdx.x) >> 5;
  int lane = threadIdx.x & 31;
  if (wid >= n) return;
  constexpr int PER = F / 32;
  float acc[PER];
#pragma unroll
  for (int p = 0; p < PER; p++) acc[p] = 0.0f;

  const int b = row_start[wid], e = row_start[wid + 1];
  for (int j = b; j < e; j++) {
    int s = csr[j];                             // broadcast load
    float w = dinv[s];                          // broadcast load
    const float* ts = t + (size_t)s * F + lane * PER;
#pragma unroll
    for (int p = 0; p < PER; p++) acc[p] += ts[p] * w;
  }

  float dv = dinv[wid];
  const float* td = t + (size_t)wid * F + lane * PER;
  float* od = out + (size_t)wid * F + lane * PER;
#pragma unroll
  for (int p = 0; p < PER; p++)
    od[p] = acc[p] * dv + td[p] * (dv * dv) + bias[lane * PER + p];
}

// ---------------- BN stats: deterministic two-stage column reduction ----------------
#define STAT_BLOCKS 256
__global__ __launch_bounds__(256)
void gcn_bn_stats(const float* __restrict__ c, float* __restrict__ pstats, int n) {
  // grid 256 x 256: stride multiple of 128 -> per-thread column invariant
  float s = 0.0f, s2 = 0.0f;
  size_t total  = (size_t)n * 128;
  size_t stride = (size_t)gridDim.x * blockDim.x;
  for (size_t idx = (size_t)blockIdx.x * blockDim.x + threadIdx.x; idx < total; idx += stride) {
    float v = c[idx];
    s += v; s2 += v * v;
  }
  pstats[(size_t)blockIdx.x * 512 + threadIdx.x]       = s;   // fixed slot, no atomics
  pstats[(size_t)blockIdx.x * 512 + 256 + threadIdx.x] = s2;
}

__global__ void gcn_bn_finalize(const float* __restrict__ pstats, const float* __restrict__ g,
                                const float* __restrict__ be, float* __restrict__ coef, int n) {
  int col = threadIdx.x;                        // 128 threads
  float s = 0.0f, s2 = 0.0f;
  for (int blk = 0; blk < STAT_BLOCKS; blk++) { // fixed order -> deterministic
    const float* p = pstats + (size_t)blk * 512;
    s  += p[col] + p[col + 128];
    s2 += p[256 + col] + p[256 + col + 128];
  }
  float invn = 1.0f / (float)n;
  float mean = s * invn;
  float var  = s2 * invn - mean * mean;
  float inv  = rsqrtf(var + 1e-5f);
  float a    = g[col] * inv;
  coef[col]       = a;                          // scale
  coef[128 + col] = be[col] - mean * a;         // shift
}

// ---------------- driver ----------------

extern "C" void kernel_launch(void* const* d_in, const int* in_sizes, int n_in,
                              void* d_out, int out_size, void* d_ws, size_t ws_size,
                              hipStream_t stream) {
  const float* x    = (const float*)d_in[0];
  const int*   eidx = (const int*)d_in[1];
  const int Nn = in_sizes[0] / 128;
  const int E  = in_sizes[1] / 2;
  const int* srcp = eidx;       // edge_index[0]
  const int* dstp = eidx + E;   // edge_index[1]

  const float* Ws[4]  = {(const float*)d_in[2],  (const float*)d_in[6],
                         (const float*)d_in[10], (const float*)d_in[14]};
  const float* bs[4]  = {(const float*)d_in[3],  (const float*)d_in[7],
                         (const float*)d_in[11], (const float*)d_in[15]};
  const float* gs[3]  = {(const float*)d_in[4],  (const float*)d_in[8],  (const float*)d_in[12]};
  const float* bes[3] = {(const float*)d_in[5],  (const float*)d_in[9],  (const float*)d_in[13]};

  const size_t Np  = ((size_t)Nn + 3) & ~(size_t)3;          // 16B-aligned element counts
  const size_t Np1 = ((size_t)Nn + 1 + 3) & ~(size_t)3;
  const size_t Ep  = ((size_t)E + 3) & ~(size_t)3;

  int* cnt       = (int*)d_ws;                               // N
  int* row_start = cnt + Np;                                 // N+1
  int* cursor    = row_start + Np1;                          // N
  int* csr       = cursor + Np;                              // E
  float* dinv    = (float*)(csr + Ep);                       // N
  float* t       = dinv + Np;                                // N*128
  float* c       = t + (size_t)Nn * 128;                     // N*128
  float* pstats  = c + (size_t)Nn * 128;                     // 256*512
  float* coef    = pstats + (size_t)STAT_BLOCKS * 512;       // 256
  (void)ws_size; (void)n_in;

  // ---- CSR build (deterministic: atomic fill + per-node sort by src) ----
  gcn_zero_i<<<(Nn + 255) / 256, 256, 0, stream>>>(cnt, Nn);
  gcn_count<<<(E + 255) / 256, 256, 0, stream>>>(dstp, cnt, E);
  gcn_scan<<<1, 1024, 0, stream>>>(cnt, row_start, cursor, Nn);
  gcn_fill<<<(E + 255) / 256, 256, 0, stream>>>(srcp, dstp, cursor, csr, E);
  gcn_sort<<<(Nn + 255) / 256, 256, 0, stream>>>(csr, row_start, Nn);
  gcn_dinv<<<(Nn + 255) / 256, 256, 0, stream>>>(cnt, dinv, Nn);

  const int tiles = (Nn + 15) / 16;
  const int gemm_grid = tiles < 512 ? tiles : 512;
  const int gather_grid = ((Nn * 32) + 255) / 256;

  // ---- layer 1: GEMM(x) -> gather conv -> BN stats/coef ----
  gcn_gemm_wmma<128, false><<<gemm_grid, 256, 0, stream>>>(x, Ws[0], nullptr, t, Nn, tiles);
  gcn_gather<128><<<gather_grid, 256, 0, stream>>>(t, dinv, csr, row_start, bs[0], c, Nn);
  gcn_bn_stats<<<STAT_BLOCKS, 256, 0, stream>>>(c, pstats, Nn);
  gcn_bn_finalize<<<1, 128, 0, stream>>>(pstats, gs[0], bes[0], coef, Nn);

  // ---- layers 2..3: GEMM with fused BN+ReLU on input -> gather -> BN stats/coef ----
  for (int l = 1; l < 3; l++) {
    gcn_gemm_wmma<128, true><<<gemm_grid, 256, 0, stream>>>(c, Ws[l], coef, t, Nn, tiles);
    gcn_gather<128><<<gather_grid, 256, 0, stream>>>(t, dinv, csr, row_start, bs[l], c, Nn);
    gcn_bn_stats<<<STAT_BLOCKS, 256, 0, stream>>>(c, pstats, Nn);
    gcn_bn_finalize<<<1, 128, 0, stream>>>(pstats, gs[l], bes[l], coef, Nn);
  }

  // ---- layer 4: GEMM(128->64) with fused BN+ReLU -> gather conv, writes d_out [N,64] ----
  float* out = (float*)d_out;
  gcn_gemm_wmma<64, true><<<gemm_grid, 128, 0, stream>>>(c, Ws[3], coef, t, Nn, tiles);
  gcn_gather<64><<<gather_grid, 256, 0, stream>>>(t, dinv, csr, row_start, bs[3], out, Nn);
}